// Model_54589034332305
// MI455X (gfx1250) — compile-verified
//
#include <hip/hip_runtime.h>
#include <hip/hip_bf16.h>
#include <math.h>

#define DEPTH  12
#define DIM    256
#define DINNER 256
#define NSTATE 256
#define DTRANK 32
#define DCONV  4
#define BATCH  2
#define SEQ    256          // L+1 after next-token concat
#define BL     (BATCH*SEQ)  // 512 rows
#define XDBL_W (DTRANK + 2*NSTATE)  // 544

#define KCHUNK 32           // K rows of W staged per TDM transfer
#define PANEL  128          // output columns per block (8 waves x 16)

typedef float        v2f   __attribute__((ext_vector_type(2)));
typedef float        v8f   __attribute__((ext_vector_type(8)));
typedef unsigned int u32x4 __attribute__((ext_vector_type(4)));
typedef int          i32x8 __attribute__((ext_vector_type(8)));
typedef int          i32x4 __attribute__((ext_vector_type(4)));

__device__ __forceinline__ float sigmoidf_(float x) { return 1.f / (1.f + __expf(-x)); }

// ---------------- concat x with broadcast next_token ----------------
__global__ void concat_kernel(const float* __restrict__ x, const float* __restrict__ nt,
                              float* __restrict__ X) {
  int i = blockIdx.x * blockDim.x + threadIdx.x;
  if (i >= BATCH * SEQ * DIM) return;
  int c = i & (DIM - 1);
  int t = (i >> 8) & (SEQ - 1);
  int b = i >> 16;
  X[i] = (t < SEQ - 1) ? x[((size_t)(b * (SEQ - 1) + t)) * DIM + c] : nt[c];
}

// ---------------- RMSNorm: one block per row (256 threads = 8 waves) ----------------
__global__ void rmsnorm_kernel(const float* __restrict__ x, const float* __restrict__ w,
                               float* __restrict__ out) {
  int row = blockIdx.x;
  int tid = threadIdx.x;
  float v  = x[(size_t)row * DIM + tid];
  float ss = v * v;
#pragma unroll
  for (int off = 16; off; off >>= 1) ss += __shfl_xor(ss, off, 32);
  __shared__ float sred[8];
  int wave = tid >> 5, lane = tid & 31;
  if (lane == 0) sred[wave] = ss;
  __syncthreads();
  float tot = 0.f;
#pragma unroll
  for (int i = 0; i < 8; i++) tot += sred[i];
  float r = rsqrtf(tot * (1.f / DIM) + 1e-5f);
  out[(size_t)row * DIM + tid] = v * r * w[tid];
}

// ---------------- WMMA GEMM with TDM-staged B panels --------------------------------
// out[M,N] = A[M,K] @ W[K,N]  (+ softplus(.+bias) epilogue, + residual add)
// grid = (M/16, ceil(N/PANEL)); block = 256 (8 waves). Each wave owns one 16x16 tile.
// Wave 0 DMA-loads each KCHUNKxPANEL chunk of W into LDS via the Tensor Data Mover;
// all waves consume it through LDS. Partial panels exit wave-uniformly (EXEC all-1s
// preserved for WMMA). mode 0: plain   mode 1: softplus(acc + bias[col])
__global__ void gemm_f32_wmma(const float* __restrict__ A, int lda,
                              const float* __restrict__ W, int ldw,
                              float* __restrict__ out, int ldo,
                              int N, int K,
                              const float* __restrict__ res, int ldr,
                              const float* __restrict__ bias, int mode) {
  __shared__ float bsm[KCHUNK * PANEL];  // 16 KB

  int wave  = threadIdx.x >> 5;
  int lane  = threadIdx.x & 31;
  int tm    = blockIdx.x;
  int panel = blockIdx.y;
  int nbase = panel * PANEL + wave * 16;
  bool active = (nbase < N);  // wave-uniform

  int m    = tm * 16 + (lane & 15);
  int nloc = wave * 16 + (lane & 15);
  int koff = (lane >> 4) << 1;  // lanes 0-15: K=0,1 ; lanes 16-31: K=2,3

  const float* arow = A + (size_t)m * lda + koff;
  v8f acc = {};

  for (int k0 = 0; k0 < K; k0 += KCHUNK) {
#if __has_builtin(__builtin_amdgcn_tensor_load_to_lds)
    if (threadIdx.x < 32) {  // wave 0 issues the tensor DMA (EXEC ignored by TDM)
      unsigned long long gaddr =
          (unsigned long long)(const void*)(W + (size_t)k0 * ldw + panel * PANEL);
      unsigned ldsbase = (unsigned)(unsigned long long)(const void*)&bsm[0];
      unsigned remN = (unsigned)(N - panel * PANEL);  // OOB columns zero-fill
      unsigned remK = (unsigned)(K - k0);
      u32x4 g0;
      g0[0] = 1u;                                   // count=1 (valid descriptor)
      g0[1] = ldsbase;                              // lds_addr
      g0[2] = (unsigned)(gaddr & 0xffffffffu);      // global_addr[31:0]
      g0[3] = (unsigned)((gaddr >> 32) & 0x1ffffffu) | 0x80000000u;  // addr[56:32]|type=2
      i32x8 g1;
      g1[0] = (int)(2u << 16);                      // data_size = 4B
      g1[1] = (int)((remN & 0xffffu) << 16);        // tensor_dim0[15:0]
      g1[2] = (int)(((remK & 0xffffu) << 16) | (remN >> 16));  // td1 lo | td0 hi
      g1[3] = (int)(((unsigned)PANEL << 16) | (remK >> 16));   // tile_dim0 | td1 hi
      g1[4] = (int)KCHUNK;                          // tile_dim1 (tile_dim2 = 0)
      g1[5] = (int)(unsigned)ldw;                   // tensor_dim0_stride[31:0]
      g1[6] = 0;                                    // stride hi bits
      g1[7] = 0;
      i32x4 g2 = {0, 0, 0, 0};                      // 2D tensor: groups 2/3 disabled
      i32x4 g3 = {0, 0, 0, 0};
      i32x8 g4 = {0, 0, 0, 0, 0, 0, 0, 0};          // extra group (clang-23 6-arg form)
      __builtin_amdgcn_tensor_load_to_lds(g0, g1, g2, g3, g4, 0);
      __builtin_amdgcn_s_wait_tensorcnt(0);
    }
#else
    for (int i = threadIdx.x; i < KCHUNK * PANEL; i += 256) {
      int r = i / PANEL, cidx = i - r * PANEL;
      int gn = panel * PANEL + cidx;
      bsm[i] = (gn < N) ? W[(size_t)(k0 + r) * ldw + gn] : 0.f;
    }
#endif
    __syncthreads();
    if (active) {
      __builtin_prefetch(arow + k0 + KCHUNK, 0, 1);
#pragma unroll
      for (int kk = 0; kk < KCHUNK; kk += 4) {
        v2f a, b;
        a.x = arow[k0 + kk];
        a.y = arow[k0 + kk + 1];
        b.x = bsm[(kk + koff) * PANEL + nloc];
        b.y = bsm[(kk + koff + 1) * PANEL + nloc];
        acc = __builtin_amdgcn_wmma_f32_16x16x4_f32(false, a, false, b, (short)0, acc,
                                                    false, false);
      }
    }
    __syncthreads();
  }

  if (!active) return;
  // C/D layout: VGPR r -> M = tm*16 + r (+8 for lanes 16-31), N = nbase + lane%16
  int col   = nbase + (lane & 15);
  int rbase = tm * 16 + ((lane >> 4) << 3);
#pragma unroll
  for (int r = 0; r < 8; r++) {
    int   row = rbase + r;
    float v   = acc[r];
    if (mode == 1) {
      v += bias[col];
      v = (v > 20.f) ? v : log1pf(__expf(v));  // softplus
    }
    if (res) v += res[(size_t)row * ldr + col];
    out[(size_t)row * ldo + col] = v;
  }
}

// ---------------- depthwise causal conv (taps=4) + bias + SiLU ----------------
__global__ void conv_silu_kernel(const float* __restrict__ uz, const float* __restrict__ cw,
                                 const float* __restrict__ cb, float* __restrict__ u) {
  int i = blockIdx.x * blockDim.x + threadIdx.x;
  if (i >= BL * DINNER) return;
  int c   = i & (DINNER - 1);
  int row = i >> 8;
  int t   = row & (SEQ - 1);
  int b   = row >> 8;
  float acc = cb[c];
#pragma unroll
  for (int j = 0; j < DCONV; j++) {
    int tt = t - (DCONV - 1) + j;
    if (tt >= 0)
      acc += cw[c * DCONV + j] * uz[((size_t)(b * SEQ + tt)) * (2 * DINNER) + c];
  }
  u[i] = acc * sigmoidf_(acc);
}

// ---------------- selective scan: one wave per (b, channel), 8 states/lane ------------
// Fused epilogue: y = (ys + u*D) * silu(z)
__global__ void scan_kernel(const float* __restrict__ delta, const float* __restrict__ xdbl,
                            const float* __restrict__ u, const float* __restrict__ uz,
                            const float* __restrict__ Alog, const float* __restrict__ Dp,
                            float* __restrict__ y) {
  int wave = threadIdx.x >> 5, lane = threadIdx.x & 31;
  int idx  = blockIdx.x * (blockDim.x >> 5) + wave;  // 0..511
  int b = idx >> 8;
  int c = idx & (DINNER - 1);

  float Av[8], h[8];
#pragma unroll
  for (int j = 0; j < 8; j++) {
    Av[j] = -__expf(Alog[(size_t)c * NSTATE + lane * 8 + j]);
    h[j]  = 0.f;
  }
  float dskip = Dp[c];

  for (int t = 0; t < SEQ; t++) {
    int   row = b * SEQ + t;
    float d   = delta[(size_t)row * DINNER + c];  // uniform across wave
    float ut  = u[(size_t)row * DINNER + c];
    float du  = d * ut;
    const float4* B4 =
        reinterpret_cast<const float4*>(xdbl + (size_t)row * XDBL_W + DTRANK + lane * 8);
    const float4* C4 = reinterpret_cast<const float4*>(
        xdbl + (size_t)row * XDBL_W + DTRANK + NSTATE + lane * 8);
    float4 b0 = B4[0], b1 = B4[1];
    float4 c0 = C4[0], c1 = C4[1];
    float  bb[8] = {b0.x, b0.y, b0.z, b0.w, b1.x, b1.y, b1.z, b1.w};
    float  cc[8] = {c0.x, c0.y, c0.z, c0.w, c1.x, c1.y, c1.z, c1.w};
    float  part  = 0.f;
#pragma unroll
    for (int j = 0; j < 8; j++) {
      h[j] = __expf(d * Av[j]) * h[j] + du * bb[j];
      part += h[j] * cc[j];
    }
#pragma unroll
    for (int off = 16; off; off >>= 1) part += __shfl_xor(part, off, 32);
    if (lane == 0) {
      float yv = part + ut * dskip;
      float z  = uz[(size_t)row * (2 * DINNER) + DINNER + c];
      y[(size_t)row * DINNER + c] = yv * (z * sigmoidf_(z));
    }
  }
}

// ---------------------------------------------------------------------------------------
extern "C" void kernel_launch(void* const* d_in, const int* in_sizes, int n_in,
                              void* d_out, int out_size, void* d_ws, size_t ws_size,
                              hipStream_t stream) {
  const float* x       = (const float*)d_in[0];
  const float* nt      = (const float*)d_in[1];
  const float* norm_w  = (const float*)d_in[2];
  const float* W_in    = (const float*)d_in[3];
  const float* conv_w  = (const float*)d_in[4];
  const float* conv_b  = (const float*)d_in[5];
  const float* W_x     = (const float*)d_in[6];
  const float* W_dt    = (const float*)d_in[7];
  const float* b_dt    = (const float*)d_in[8];
  const float* A_log   = (const float*)d_in[9];
  const float* D_skip  = (const float*)d_in[10];
  const float* W_out   = (const float*)d_in[11];
  const float* final_w = (const float*)d_in[12];

  float* X     = (float*)d_ws;               // 512*256
  float* H     = X + (size_t)BL * DIM;       // 512*256
  float* UZ    = H + (size_t)BL * DIM;       // 512*512
  float* U     = UZ + (size_t)BL * 2 * DINNER;
  float* XDBL  = U + (size_t)BL * DINNER;    // 512*544
  float* DELTA = XDBL + (size_t)BL * XDBL_W;
  float* Y     = DELTA + (size_t)BL * DINNER;

  concat_kernel<<<(BL * DIM + 255) / 256, 256, 0, stream>>>(x, nt, X);

  for (int d = 0; d < DEPTH; d++) {
    const float* nw  = norm_w + (size_t)d * DIM;
    const float* Wi  = W_in + (size_t)d * DIM * 2 * DINNER;
    const float* cw  = conv_w + (size_t)d * DINNER * DCONV;
    const float* cb  = conv_b + (size_t)d * DINNER;
    const float* Wx  = W_x + (size_t)d * DINNER * XDBL_W;
    const float* Wdt = W_dt + (size_t)d * DTRANK * DINNER;
    const float* bdt = b_dt + (size_t)d * DINNER;
    const float* Al  = A_log + (size_t)d * DINNER * NSTATE;
    const float* Dpp = D_skip + (size_t)d * DINNER;
    const float* Wo  = W_out + (size_t)d * DINNER * DIM;

    // h = rmsnorm(x) * nw
    rmsnorm_kernel<<<BL, DIM, 0, stream>>>(X, nw, H);

    // UZ = H @ Wi  (512 x 512 x 256)
    gemm_f32_wmma<<<dim3(BL / 16, (2 * DINNER + PANEL - 1) / PANEL), 256, 0, stream>>>(
        H, DIM, Wi, 2 * DINNER, UZ, 2 * DINNER, 2 * DINNER, DIM, nullptr, 0, nullptr, 0);

    // u = silu(causal_conv(UZ[:, :256]) + cb)
    conv_silu_kernel<<<(BL * DINNER + 255) / 256, 256, 0, stream>>>(UZ, cw, cb, U);

    // XDBL = U @ Wx  (512 x 544 x 256)
    gemm_f32_wmma<<<dim3(BL / 16, (XDBL_W + PANEL - 1) / PANEL), 256, 0, stream>>>(
        U, DINNER, Wx, XDBL_W, XDBL, XDBL_W, XDBL_W, DINNER, nullptr, 0, nullptr, 0);

    // DELTA = softplus(XDBL[:, :32] @ Wdt + bdt)  (512 x 256 x 32)
    gemm_f32_wmma<<<dim3(BL / 16, (DINNER + PANEL - 1) / PANEL), 256, 0, stream>>>(
        XDBL, XDBL_W, Wdt, DINNER, DELTA, DINNER, DINNER, DTRANK, nullptr, 0, bdt, 1);

    // selective scan + D-skip + SiLU(z) gate  (512 waves)
    scan_kernel<<<(BATCH * DINNER) / 8, 256, 0, stream>>>(DELTA, XDBL, U, UZ, Al, Dpp, Y);

    // X = X + Y @ Wo  (512 x 256 x 256, residual add in-place)
    gemm_f32_wmma<<<dim3(BL / 16, (DIM + PANEL - 1) / PANEL), 256, 0, stream>>>(
        Y, DINNER, Wo, DIM, X, DIM, DIM, DINNER, X, DIM, nullptr, 0);
  }

  // final rmsnorm -> d_out
  rmsnorm_kernel<<<BL, DIM, 0, stream>>>(X, final_w, (float*)d_out);
}